// HMAUNetLoss_43190191129260
// MI455X (gfx1250) — compile-verified
//
#include <hip/hip_runtime.h>
#include <math.h>
#include <stdint.h>

// Problem dims (fixed by reference: B=16, C=1, H=W=512)
#define HH   512
#define WW   512
#define BATCH 16
#define HWN  (HH*WW)
#define NTOT (BATCH*HWN)

static constexpr float A_      = 0.955f;    // axial chamfer cost
static constexpr float BD_     = 1.3693f;   // diagonal chamfer cost
static constexpr float INF_    = 1e6f;
static constexpr float ALPHA_  = 0.25f;
static constexpr float SMOOTH_ = 1e-6f;

typedef __attribute__((ext_vector_type(2))) float v2f;
typedef __attribute__((ext_vector_type(8))) float v8f;

// ---------------------------------------------------------------------------
// Async global->LDS row prefetch (CDNA5 GLOBAL_LOAD_ASYNC_TO_LDS_*, ASYNCcnt).
// Builtin params are typed addrspace pointers: (AS1 char*/int*, AS3 char*/int*,
// imm offset, imm cpol).  Guarded: falls back to register software-pipelining.
// ---------------------------------------------------------------------------
#if defined(__has_builtin)
#if __has_builtin(__builtin_amdgcn_global_load_async_to_lds_b32) && \
    __has_builtin(__builtin_amdgcn_global_load_async_to_lds_b8)  && \
    __has_builtin(__builtin_amdgcn_s_wait_asynccnt)
#define USE_ASYNC_LDS 1
#endif
#endif
#ifndef USE_ASYNC_LDS
#define USE_ASYNC_LDS 0
#endif

#if USE_ASYNC_LDS
typedef __attribute__((address_space(1))) char GlbChar;
typedef __attribute__((address_space(1))) int  GlbInt;
typedef __attribute__((address_space(3))) char LdsChar;
typedef __attribute__((address_space(3))) int  LdsInt;
__device__ inline void async_row_b8(const void* g, const void* l) {
  __builtin_amdgcn_global_load_async_to_lds_b8(
      (GlbChar*)(unsigned long long)(uintptr_t)g,
      (LdsChar*)(unsigned int)(uintptr_t)l, 0, 0);
}
__device__ inline void async_row_b32(const void* g, const void* l) {
  __builtin_amdgcn_global_load_async_to_lds_b32(
      (GlbInt*)(unsigned long long)(uintptr_t)g,
      (LdsInt*)(unsigned int)(uintptr_t)l, 0, 0);
}
#endif

// ---------------------------------------------------------------------------
// Exact wave32 sum-reduction via V_WMMA_F32_16X16X4_F32 (f32 in, f32 acc).
// B = all-ones => D[m][n] = sum_k A[m][k]; summing a lane's 8 D VGPRs covers
// m=0..7 (lanes<16) / m=8..15 (lanes>=16) of one column; lane0+lane16 is the
// full wave sum, computed entirely in f32.
// ---------------------------------------------------------------------------
__device__ inline float wave_reduce_sum_wmma(float v) {
  v2f a; a.x = v;    a.y = 0.0f;
  v2f b; b.x = 1.0f; b.y = 1.0f;
  v8f c = {};
  c = __builtin_amdgcn_wmma_f32_16x16x4_f32(false, a, false, b, (short)0, c,
                                            false, false);
  float s = c[0] + c[1] + c[2] + c[3] + c[4] + c[5] + c[6] + c[7];
  return __shfl(s, 0, 32) + __shfl(s, 16, 32);
}

// ---------------------------------------------------------------------------
// Kernel 1: seed codes. bit0 = mask (t>0.5), bit1 = boundary (dil && inv).
// reduce_window init=0 => out-of-bounds contributes 0 to both maxpools.
// ---------------------------------------------------------------------------
__global__ void __launch_bounds__(256)
k_seeds(const int* __restrict__ target, unsigned char* __restrict__ code,
        int* __restrict__ hasFG, int* __restrict__ hasB) {
  int idx = blockIdx.x * 256 + threadIdx.x;
  int b = idx / HWN;
  int r = idx - b * HWN;
  int i = r / WW, j = r - i * WW;
  const int* tb = target + b * HWN;

  int m = tb[r] > 0 ? 1 : 0;
  int dil = 0, inv = 0;
  #pragma unroll
  for (int di = -1; di <= 1; ++di) {
    int ii = i + di;
    if (ii < 0 || ii >= HH) continue;
    #pragma unroll
    for (int dj = -1; dj <= 1; ++dj) {
      int jj = j + dj;
      if (jj < 0 || jj >= WW) continue;
      int mm = tb[ii * WW + jj] > 0 ? 1 : 0;
      dil |= mm;
      inv |= (1 - mm);
    }
  }
  int boundary = dil & inv;
  code[idx] = (unsigned char)(m | (boundary << 1));

  // HWN % 32 == 0 -> waves never straddle samples
  unsigned long long bm = __ballot(m);
  unsigned long long bb = __ballot(boundary);
  if ((threadIdx.x & 31) == 0) {
    if (bm) atomicOr(&hasFG[b], 1);
    if (bb) atomicOr(&hasB[b], 1);
  }
}

// ---------------------------------------------------------------------------
// Kernel 2: forward chamfer sweep.  One block (512 thr, 16 waves) per sample.
//   m[j]   = min(d0[j], prev[j]+A, prev[j-1]+B, prev[j+1]+B)
//   cur[j] = A*j + cummin_{k<=j}(m[k] - A*k)
// 2 barriers/row: double-buffered sPrev + redundant per-wave cross-scan.
// Next row's code bytes async-prefetched into LDS (per-wave asynccnt wait).
// ---------------------------------------------------------------------------
__global__ void __launch_bounds__(512)
k_chamfer_fwd(const unsigned char* __restrict__ code,
              const int* __restrict__ hasB, float* __restrict__ dist) {
  __shared__ float sPrev[2][WW];
  __shared__ float sWaveTot[16];
#if USE_ASYNC_LDS
  __shared__ unsigned char sCode[2][WW];
#endif
  const int b = blockIdx.x;
  const int j = threadIdx.x;
  const int lane = j & 31, wid = j >> 5;
  const bool useB = hasB[b] != 0;
  const unsigned char* cb = code + b * HWN;
  float* db = dist + b * HWN;
  const float aj = A_ * (float)j;

  sPrev[0][j] = INF_;
#if USE_ASYNC_LDS
  async_row_b8(&cb[j], &sCode[0][j]);
#else
  unsigned char cnext = cb[j];
#endif
  __syncthreads();
  int pb = 0;

  for (int i = 0; i < HH; ++i) {
#if USE_ASYNC_LDS
    const int rb = i & 1;
    if (i < HH - 1) {                 // issue next row, wait for current
      async_row_b8(&cb[(i + 1) * WW + j], &sCode[rb ^ 1][j]);
      __builtin_amdgcn_s_wait_asynccnt(1);
    } else {
      __builtin_amdgcn_s_wait_asynccnt(0);
    }
    unsigned char c = sCode[rb][j];   // own column -> per-wave wait suffices
#else
    unsigned char c = cnext;
    if (i < HH - 1) cnext = cb[(i + 1) * WW + j];
#endif
    bool seed = useB ? ((c & 2) != 0) : ((c & 1) == 0);
    float dval = seed ? 0.0f : INF_;
    float up = sPrev[pb][j] + A_;
    float ul = ((j > 0)      ? sPrev[pb][j - 1] : INF_) + BD_;
    float ur = ((j < WW - 1) ? sPrev[pb][j + 1] : INF_) + BD_;
    float v = fminf(fminf(dval, up), fminf(ul, ur)) - aj;

    // inclusive prefix-min inside wave
    #pragma unroll
    for (int off = 1; off < 32; off <<= 1) {
      float t = __shfl_up(v, off, 32);
      if (lane >= off) v = fminf(v, t);
    }
    if (lane == 31) sWaveTot[wid] = v;
    __syncthreads();                                    // (a)

    // every wave redundantly scans the 16 wave totals (prefix-min)
    float wv = (lane < 16) ? sWaveTot[lane] : INF_;
    #pragma unroll
    for (int off = 1; off < 16; off <<= 1) {
      float t = __shfl_up(wv, off, 32);
      if (lane >= off) wv = fminf(wv, t);
    }
    if (wid > 0) v = fminf(v, __shfl(wv, wid - 1, 32)); // exclusive prefix

    float curv = aj + v;
    db[i * WW + j] = curv;
    sPrev[pb ^ 1][j] = curv;
    pb ^= 1;
    __syncthreads();                                    // (b)
  }
}

// ---------------------------------------------------------------------------
// Kernel 3: backward sweep == chamfer pass on the doubly-flipped grid:
//   m[j]   = min(d1[j], next[j]+A, next[j-1]+B, next[j+1]+B)   (rows H-1..0)
//   cur[j] = -A*j + suffixmin_{k>=j}(m[k] + A*k)
// Also produces the per-sample max. Dist rows async-prefetched into LDS.
// ---------------------------------------------------------------------------
__global__ void __launch_bounds__(512)
k_chamfer_bwd(float* __restrict__ dist, float* __restrict__ maxd) {
  __shared__ float sPrev[2][WW];
  __shared__ float sWaveTot[16];
  __shared__ float sMax[16];
#if USE_ASYNC_LDS
  __shared__ float sRowD[2][WW];
#endif
  const int b = blockIdx.x;
  const int j = threadIdx.x;
  const int lane = j & 31, wid = j >> 5;
  float* db = dist + b * HWN;
  const float aj = A_ * (float)j;

  sPrev[0][j] = INF_;
#if USE_ASYNC_LDS
  async_row_b32(&db[(HH - 1) * WW + j], &sRowD[0][j]);
#else
  float dnext = db[(HH - 1) * WW + j];
#endif
  __syncthreads();
  int pb = 0;
  float lmax = 0.0f;

  for (int i = HH - 1; i >= 0; --i) {
#if USE_ASYNC_LDS
    const int rb = (HH - 1 - i) & 1;
    if (i > 0) {
      async_row_b32(&db[(i - 1) * WW + j], &sRowD[rb ^ 1][j]);
      __builtin_amdgcn_s_wait_asynccnt(1);
    } else {
      __builtin_amdgcn_s_wait_asynccnt(0);
    }
    float dval = sRowD[rb][j];
#else
    float dval = dnext;
    if (i > 0) dnext = db[(i - 1) * WW + j];
#endif
    float dn = sPrev[pb][j] + A_;
    float dl = ((j > 0)      ? sPrev[pb][j - 1] : INF_) + BD_;
    float dr = ((j < WW - 1) ? sPrev[pb][j + 1] : INF_) + BD_;
    float v = fminf(fminf(dval, dn), fminf(dl, dr)) + aj;

    // inclusive suffix-min inside wave
    #pragma unroll
    for (int off = 1; off < 32; off <<= 1) {
      float t = __shfl_down(v, off, 32);
      if (lane + off < 32) v = fminf(v, t);
    }
    if (lane == 0) sWaveTot[wid] = v;
    __syncthreads();                                    // (a)

    // every wave redundantly suffix-scans the 16 wave totals
    float wv = (lane < 16) ? sWaveTot[lane] : INF_;
    #pragma unroll
    for (int off = 1; off < 16; off <<= 1) {
      float t = __shfl_down(wv, off, 32);
      if (lane + off < 16) wv = fminf(wv, t);
    }
    if (wid < 15) v = fminf(v, __shfl(wv, wid + 1, 32));

    float curv = v - aj;
    db[i * WW + j] = curv;
    lmax = fmaxf(lmax, curv);
    sPrev[pb ^ 1][j] = curv;
    pb ^= 1;
    __syncthreads();                                    // (b)
  }

  #pragma unroll
  for (int off = 16; off > 0; off >>= 1)
    lmax = fmaxf(lmax, __shfl_down(lmax, off, 32));
  if (lane == 0) sMax[wid] = lmax;
  __syncthreads();
  if (j == 0) {
    float mx = sMax[0];
    for (int w = 1; w < 16; ++w) mx = fmaxf(mx, sMax[w]);
    maxd[b] = mx;
  }
}

// ---------------------------------------------------------------------------
// Per-element loss terms
// ---------------------------------------------------------------------------
__device__ inline void loss_elem(float x, float t, float d, bool fg,
                                 float invmx, float& aF, float& aB, float& aI,
                                 float& aP, float& aT) {
  float p = 1.0f / (1.0f + expf(-x));
  float l1pe = log1pf(expf(-fabsf(x)));
  float sp_pos = fmaxf(x, 0.0f) + l1pe;     // softplus(x)  = -logsig(-x)
  float sp_neg = fmaxf(-x, 0.0f) + l1pe;    // softplus(-x) = -logsig(x)
  float bce = t * sp_neg + (1.0f - t) * sp_pos;
  float pt = (t == 1.0f) ? p : (1.0f - p);
  float at = (t == 1.0f) ? ALPHA_ : (1.0f - ALPHA_);
  float om = 1.0f - pt;
  aF += at * om * om * bce;
  float dnorm = fg ? (d * invmx) : 1.0f;
  aB += (t * (1.0f - p) + (1.0f - t) * p) * (1.0f + dnorm);
  aI += p * t;
  aP += p;
  aT += t;
}

// ---------------------------------------------------------------------------
// Kernel 4: fused loss partials. 1024 blocks (64/sample) x 256 threads.
// float4/int4 (b128) streaming loads + global_prefetch; exact f32-WMMA wave
// reduction; deterministic per-block partials.
// partial[blk*5 + {0:focal, 1:bnd, 2:inter, 3:psum, 4:tsum}]
// ---------------------------------------------------------------------------
__global__ void __launch_bounds__(256)
k_loss_partial(const float* __restrict__ pred, const int* __restrict__ target,
               const float* __restrict__ dist, const float* __restrict__ maxd,
               const int* __restrict__ hasFG, float* __restrict__ partial) {
  const int blk = blockIdx.x;
  const int b = blk >> 6;
  const int sub = blk & 63;
  const int tid = threadIdx.x;
  const int lane = tid & 31, wid = tid >> 5;

  const bool fg = hasFG[b] != 0;
  const float mx = maxd[b];
  const float invmx = (mx > 0.0f) ? 1.0f / fmaxf(mx, 1e-12f) : 1.0f;

  const int base4 = (b * HWN + sub * 4096) >> 2;        // 16B-aligned chunks
  const float4* pred4 = reinterpret_cast<const float4*>(pred) + base4;
  const int4*   tgt4  = reinterpret_cast<const int4*>(target) + base4;
  const float4* dist4 = reinterpret_cast<const float4*>(dist) + base4;

  float aF = 0.f, aB = 0.f, aI = 0.f, aP = 0.f, aT = 0.f;
  #pragma unroll
  for (int k = tid; k < 1024; k += 256) {
    __builtin_prefetch(&pred4[k + 256], 0, 0);          // global_prefetch_b8
    __builtin_prefetch(&dist4[k + 256], 0, 0);
    float4 x4 = pred4[k];
    int4   t4 = tgt4[k];
    float4 d4 = dist4[k];
    loss_elem(x4.x, (float)t4.x, d4.x, fg, invmx, aF, aB, aI, aP, aT);
    loss_elem(x4.y, (float)t4.y, d4.y, fg, invmx, aF, aB, aI, aP, aT);
    loss_elem(x4.z, (float)t4.z, d4.z, fg, invmx, aF, aB, aI, aP, aT);
    loss_elem(x4.w, (float)t4.w, d4.w, fg, invmx, aF, aB, aI, aP, aT);
  }

  __shared__ float red[8 * 5];
  float rF  = wave_reduce_sum_wmma(aF);
  float rB  = wave_reduce_sum_wmma(aB);
  float rI  = wave_reduce_sum_wmma(aI);
  float rP  = wave_reduce_sum_wmma(aP);
  float rT  = wave_reduce_sum_wmma(aT);
  if (lane == 0) {
    red[wid * 5 + 0] = rF; red[wid * 5 + 1] = rB; red[wid * 5 + 2] = rI;
    red[wid * 5 + 3] = rP; red[wid * 5 + 4] = rT;
  }
  __syncthreads();
  if (tid == 0) {
    float s0 = 0, s1 = 0, s2 = 0, s3 = 0, s4 = 0;
    for (int w = 0; w < 8; ++w) {
      s0 += red[w * 5 + 0]; s1 += red[w * 5 + 1]; s2 += red[w * 5 + 2];
      s3 += red[w * 5 + 3]; s4 += red[w * 5 + 4];
    }
    float* pp = partial + blk * 5;
    pp[0] = s0; pp[1] = s1; pp[2] = s2; pp[3] = s3; pp[4] = s4;
  }
}

// ---------------------------------------------------------------------------
// Kernel 5: deterministic final combine -> out = {total, focal, dice, bnd, iou}
// ---------------------------------------------------------------------------
__global__ void __launch_bounds__(64)
k_final(const float* __restrict__ partial, const float* __restrict__ log_vars,
        float* __restrict__ out) {
  __shared__ float sDice[16], sIou[16];
  __shared__ float sFocal, sBnd;
  const int tid = threadIdx.x;
  if (tid < 16) {
    float inter = 0, ps = 0, ts = 0;
    for (int k = 0; k < 64; ++k) {
      const float* pp = partial + (tid * 64 + k) * 5;
      inter += pp[2]; ps += pp[3]; ts += pp[4];
    }
    float total = ps + ts;
    sDice[tid] = (2.0f * inter + SMOOTH_) / (total + SMOOTH_);
    sIou[tid]  = (inter + SMOOTH_) / ((total - inter) + SMOOTH_);
  } else if (tid == 16) {
    float s = 0;
    for (int i = 0; i < 1024; ++i) s += partial[i * 5 + 0];
    sFocal = s;
  } else if (tid == 17) {
    float s = 0;
    for (int i = 0; i < 1024; ++i) s += partial[i * 5 + 1];
    sBnd = s;
  }
  __syncthreads();
  if (tid == 0) {
    float dsum = 0, isum = 0;
    for (int b = 0; b < 16; ++b) { dsum += sDice[b]; isum += sIou[b]; }
    const float inv_n = 1.0f / (float)NTOT;
    float focal = sFocal * inv_n;
    float dice  = 1.0f - dsum * (1.0f / 16.0f);
    float bnd   = sBnd * inv_n;
    float iou   = 1.0f - isum * (1.0f / 16.0f);
    float lv0 = log_vars[0], lv1 = log_vars[1], lv2 = log_vars[2], lv3 = log_vars[3];
    float total = expf(-lv0) * focal + lv0 + expf(-lv1) * dice + lv1 +
                  expf(-lv2) * bnd  + lv2 + expf(-lv3) * iou  + lv3;
    out[0] = total; out[1] = focal; out[2] = dice; out[3] = bnd; out[4] = iou;
  }
}

__global__ void k_init_flags(int* __restrict__ flags) {
  if (threadIdx.x < 32) flags[threadIdx.x] = 0;
}

// ---------------------------------------------------------------------------
extern "C" void kernel_launch(void* const* d_in, const int* in_sizes, int n_in,
                              void* d_out, int out_size, void* d_ws,
                              size_t ws_size, hipStream_t stream) {
  const float* pred     = (const float*)d_in[0];
  const int*   target   = (const int*)d_in[1];
  const float* log_vars = (const float*)d_in[2];
  float* out = (float*)d_out;

  // workspace layout
  float* dist = (float*)d_ws;                               // NTOT f32
  unsigned char* code = (unsigned char*)(dist + NTOT);      // NTOT u8
  int* flags = (int*)(code + NTOT);                         // hasFG[16], hasB[16]
  float* maxd = (float*)(flags + 32);                       // 16 f32
  float* partial = maxd + 16;                               // 1024*5 f32
  int* hasFG = flags;
  int* hasB  = flags + 16;

  k_init_flags<<<dim3(1), dim3(64), 0, stream>>>(flags);
  k_seeds<<<dim3(NTOT / 256), dim3(256), 0, stream>>>(target, code, hasFG, hasB);
  k_chamfer_fwd<<<dim3(BATCH), dim3(512), 0, stream>>>(code, hasB, dist);
  k_chamfer_bwd<<<dim3(BATCH), dim3(512), 0, stream>>>(dist, maxd);
  k_loss_partial<<<dim3(BATCH * 64), dim3(256), 0, stream>>>(pred, target, dist,
                                                             maxd, hasFG, partial);
  k_final<<<dim3(1), dim3(64), 0, stream>>>(partial, log_vars, out);
}